// SharedPerPositionSAE_3805341024316
// MI455X (gfx1250) — compile-verified
//
#include <hip/hip_runtime.h>

// Problem constants (match reference)
#define D_IN   768
#define D_SAE  16384
#define NROWS  4096   // B*T = 32*128
#define KTOP   32

typedef __attribute__((ext_vector_type(2))) float v2f;
typedef __attribute__((ext_vector_type(8))) float v8f;

// ---------------------------------------------------------------------------
// Kernel 1: encoder GEMM with fp32 WMMA (V_WMMA_F32_16X16X4_F32).
//   pre[n,s] = sum_d (x[n,d]-b_dec[d]) * W_enc[s,d] + b_enc[s]
// Block: 256 threads = 8 waves. Block tile: 16 rows (M) x 512 cols (N).
// Each wave: 16x64 output tile (four v8f accumulators), K-loop in steps of 4.
// Fragment layout (ISA 7.12.2, 32-bit A 16x4): lanes 0-15 hold K=k,k+1;
// lanes 16-31 hold K=k+2,k+3. The +2 K-offset for the high half-wave is baked
// into the load pointers, so the inner loop has NO per-iteration selects:
//   1x ds_load_b64 (A) + 4x global_load_b64 (B) + 4x v_wmma per k-step.
// A (xc tile, 16x768) staged in LDS, padded stride 772 (bank-conflict free).
// ---------------------------------------------------------------------------
__launch_bounds__(256)
__global__ void sae_encoder_wmma(const float* __restrict__ x,
                                 const float* __restrict__ b_dec,
                                 const float* __restrict__ W_enc,
                                 const float* __restrict__ b_enc,
                                 float* __restrict__ pre /* [NROWS, D_SAE] */)
{
    __shared__ float sA[16][772];   // 16 x 768 + pad; 49,408 B

    const int tid  = threadIdx.x;
    const int bn   = blockIdx.x;    // col block: 512 cols
    const int bm   = blockIdx.y;    // row block: 16 rows
    const int row0 = bm * 16;

    // Stage xc = x - b_dec for this 16-row tile (coalesced: consecutive d).
    for (int i = tid; i < 16 * D_IN; i += 256) {
        int r = i / D_IN;
        int d = i - r * D_IN;
        sA[r][d] = x[(size_t)(row0 + r) * D_IN + d] - b_dec[d];
    }
    __syncthreads();

    const int  wave = tid >> 5;
    const int  lane = tid & 31;
    const int  lm   = lane & 15;
    const int  ks   = (lane >= 16) ? 2 : 0;     // half-wave K offset
    const int  colw = bn * 512 + wave * 64;     // this wave's 64 columns

    // Per-lane base pointers with the half-wave K offset baked in.
    const float* __restrict__ ap = &sA[lm][ks];
    const float* __restrict__ w0 = W_enc + (size_t)(colw + 0  + lm) * D_IN + ks;
    const float* __restrict__ w1 = W_enc + (size_t)(colw + 16 + lm) * D_IN + ks;
    const float* __restrict__ w2 = W_enc + (size_t)(colw + 32 + lm) * D_IN + ks;
    const float* __restrict__ w3 = W_enc + (size_t)(colw + 48 + lm) * D_IN + ks;

    v8f c0 = {}, c1 = {}, c2 = {}, c3 = {};

    for (int k = 0; k < D_IN; k += 4) {
        v2f a  = *(const v2f*)(ap + k);   // ds_load_b64
        v2f b0 = *(const v2f*)(w0 + k);   // global_load_b64 (unique 8B/lane)
        v2f b1 = *(const v2f*)(w1 + k);
        v2f b2 = *(const v2f*)(w2 + k);
        v2f b3 = *(const v2f*)(w3 + k);

        c0 = __builtin_amdgcn_wmma_f32_16x16x4_f32(false, a, false, b0,
                                                   (short)0, c0, false, false);
        c1 = __builtin_amdgcn_wmma_f32_16x16x4_f32(false, a, false, b1,
                                                   (short)0, c1, false, false);
        c2 = __builtin_amdgcn_wmma_f32_16x16x4_f32(false, a, false, b2,
                                                   (short)0, c2, false, false);
        c3 = __builtin_amdgcn_wmma_f32_16x16x4_f32(false, a, false, b3,
                                                   (short)0, c3, false, false);
    }

    // C/D layout: VGPR v, lanes 0-15 -> (M=v, N=lane); lanes 16-31 -> (M=v+8).
    const int rb = row0 + ((lane >= 16) ? 8 : 0);
    const float be0 = b_enc[colw + 0  + lm];
    const float be1 = b_enc[colw + 16 + lm];
    const float be2 = b_enc[colw + 32 + lm];
    const float be3 = b_enc[colw + 48 + lm];
#pragma unroll
    for (int v = 0; v < 8; ++v) {
        float* prow = pre + (size_t)(rb + v) * D_SAE + colw + lm;
        prow[0]  = c0[v] + be0;   // lanes 0-15 of each half coalesce (64B)
        prow[16] = c1[v] + be1;
        prow[32] = c2[v] + be2;
        prow[48] = c3[v] + be3;
    }
}

// ---------------------------------------------------------------------------
// Kernel 2: per-row top-32 (+ReLU), in place on the pre/z buffer.
// One block (256 threads) per row. Each thread holds 64 elements in registers
// (col = j*256 + tid), tracks selected elements in a 64-bit mask, and the
// block runs 32 argmax rounds (wave shuffle reduce + tiny LDS cross-wave).
// Tie-break: lowest index (matches jax.lax.top_k).
// ---------------------------------------------------------------------------
__launch_bounds__(256)
__global__ void sae_topk(float* __restrict__ pre /* in: pre, out: sparse z */,
                         int*   __restrict__ ws_idx,
                         float* __restrict__ ws_val)
{
    const int n   = blockIdx.x;
    const int tid = threadIdx.x;
    float* __restrict__ row = pre + (size_t)n * D_SAE;

    float v[64];
#pragma unroll
    for (int j = 0; j < 64; ++j) v[j] = row[j * 256 + tid];

    unsigned long long sel = 0ull;

    __shared__ float s_wv[8];
    __shared__ int   s_wi[8];
    __shared__ float s_bv;
    __shared__ int   s_bi;

    const int lane = tid & 31;
    const int wave = tid >> 5;

    for (int r = 0; r < KTOP; ++r) {
        // local argmax over unselected elements
        float best = -INFINITY;
        int   bidx = 0x7fffffff;
#pragma unroll
        for (int j = 0; j < 64; ++j) {
            if (!((sel >> j) & 1ull)) {
                float val = v[j];
                int   col = j * 256 + tid;
                if (val > best || (val == best && col < bidx)) {
                    best = val;
                    bidx = col;
                }
            }
        }
        // wave32 shuffle reduce (argmax, lowest-index tie-break)
        for (int off = 16; off > 0; off >>= 1) {
            float ov = __shfl_xor(best, off, 32);
            int   oi = __shfl_xor(bidx, off, 32);
            if (ov > best || (ov == best && oi < bidx)) {
                best = ov;
                bidx = oi;
            }
        }
        if (lane == 0) { s_wv[wave] = best; s_wi[wave] = bidx; }
        __syncthreads();
        if (tid == 0) {
            float bb = s_wv[0];
            int   bi = s_wi[0];
            for (int w = 1; w < 8; ++w) {
                float ov = s_wv[w];
                int   oi = s_wi[w];
                if (ov > bb || (ov == bb && oi < bi)) { bb = ov; bi = oi; }
            }
            s_bv = bb;
            s_bi = bi;
            ws_idx[n * KTOP + r] = bi;
            ws_val[n * KTOP + r] = fmaxf(bb, 0.0f);   // relu'd value for decoder
        }
        __syncthreads();
        const int widx = s_bi;
        if ((widx & 255) == tid) sel |= 1ull << (widx >> 8);
        // next round's writes to s_wv/s_bi are gated by the barrier above
    }

    // write sparse z: relu(v) where selected, else 0 (coalesced)
#pragma unroll
    for (int j = 0; j < 64; ++j) {
        float o = ((sel >> j) & 1ull) ? fmaxf(v[j], 0.0f) : 0.0f;
        row[j * 256 + tid] = o;
    }
}

// ---------------------------------------------------------------------------
// Kernel 3: sparse decoder + per-row squared error.
//   x_hat[n,d] = b_dec[d] + sum_j val_j * W_dec[d, idx_j]
// W_dec (50 MB) is fully L2-resident; the gather stays on-chip.
// ---------------------------------------------------------------------------
__launch_bounds__(256)
__global__ void sae_decoder(const float* __restrict__ x,
                            const float* __restrict__ b_dec,
                            const float* __restrict__ W_dec, /* [D_IN, D_SAE] */
                            const int*   __restrict__ ws_idx,
                            const float* __restrict__ ws_val,
                            float* __restrict__ x_hat,
                            float* __restrict__ partial)
{
    const int n   = blockIdx.x;
    const int tid = threadIdx.x;

    __shared__ int   si[KTOP];
    __shared__ float sv[KTOP];
    __shared__ float red[256];

    if (tid < KTOP) {
        si[tid] = ws_idx[n * KTOP + tid];
        sv[tid] = ws_val[n * KTOP + tid];
    }
    __syncthreads();

    float lsum = 0.0f;
    for (int d = tid; d < D_IN; d += 256) {
        const float* __restrict__ wr = W_dec + (size_t)d * D_SAE;
        float acc = b_dec[d];
#pragma unroll
        for (int j = 0; j < KTOP; ++j) acc += sv[j] * wr[si[j]];
        x_hat[(size_t)n * D_IN + d] = acc;
        float diff = acc - x[(size_t)n * D_IN + d];
        lsum += diff * diff;
    }

    red[tid] = lsum;
    __syncthreads();
    for (int s = 128; s > 0; s >>= 1) {
        if (tid < s) red[tid] += red[tid + s];
        __syncthreads();
    }
    if (tid == 0) partial[n] = red[0];
}

// ---------------------------------------------------------------------------
// Kernel 4: deterministic final reduction of the loss.
// ---------------------------------------------------------------------------
__launch_bounds__(256)
__global__ void sae_reduce(const float* __restrict__ partial,
                           float* __restrict__ recon)
{
    __shared__ float red[256];
    const int tid = threadIdx.x;
    float s = 0.0f;
    for (int i = tid; i < NROWS; i += 256) s += partial[i];
    red[tid] = s;
    __syncthreads();
    for (int st = 128; st > 0; st >>= 1) {
        if (tid < st) red[tid] += red[tid + st];
        __syncthreads();
    }
    if (tid == 0) recon[0] = red[0] * (1.0f / (float)NROWS);
}

// ---------------------------------------------------------------------------
// Launch: out = [recon(1) | x_hat(4096*768) | z(4096*16384)] fp32.
// pre is computed directly into the z region, then sparsified in place.
// ---------------------------------------------------------------------------
extern "C" void kernel_launch(void* const* d_in, const int* in_sizes, int n_in,
                              void* d_out, int out_size, void* d_ws, size_t ws_size,
                              hipStream_t stream)
{
    const float* x     = (const float*)d_in[0];
    const float* b_dec = (const float*)d_in[1];
    const float* W_enc = (const float*)d_in[2];
    const float* b_enc = (const float*)d_in[3];
    const float* W_dec = (const float*)d_in[4];
    // d_in[5] is k == 32 (compile-time constant here)

    float* out   = (float*)d_out;
    float* recon = out;
    float* x_hat = out + 1;
    float* z     = out + 1 + (size_t)NROWS * D_IN;

    int*   ws_idx  = (int*)d_ws;
    float* ws_val  = (float*)((char*)d_ws + (size_t)NROWS * KTOP * sizeof(int));
    float* partial = (float*)((char*)d_ws + 2ull * NROWS * KTOP * sizeof(int));

    dim3 egrid(D_SAE / 512, NROWS / 16);  // (32, 256)
    sae_encoder_wmma<<<egrid, 256, 0, stream>>>(x, b_dec, W_enc, b_enc, z);
    sae_topk<<<NROWS, 256, 0, stream>>>(z, ws_idx, ws_val);
    sae_decoder<<<NROWS, 256, 0, stream>>>(x, b_dec, W_dec, ws_idx, ws_val,
                                           x_hat, partial);
    sae_reduce<<<1, 256, 0, stream>>>(partial, recon);
}